// TopologicalLoss_42674795053619
// MI455X (gfx1250) — compile-verified
//
#include <hip/hip_runtime.h>

typedef float v2f __attribute__((ext_vector_type(2)));
typedef float v4f __attribute__((ext_vector_type(4)));
typedef float v8f __attribute__((ext_vector_type(8)));

// Reduce one float per lane across a wave32 using V_WMMA_F32_16X16X4_F32.
// A: VGPR0 = per-lane partial (covers (M,K) pairs {(m,0)} u {(m,2)}), VGPR1 = 0.
// B: all ones. => D[m,n] = p[m] + p[m+16].
// C/D layout: VGPR r holds rowsum(r) on lanes 0-15 and rowsum(r+8) on lanes 16-31,
// so the in-lane sum of the 8 D VGPRs gives half-totals; one xor-16 permute finishes.
// Must be called with full EXEC (all 32 lanes active). Returns total in every lane.
__device__ __forceinline__ float wave_sum32(float v) {
    v2f a; a[0] = v;    a[1] = 0.0f;
    v2f b; b[0] = 1.0f; b[1] = 1.0f;
    v8f c = {};
    c = __builtin_amdgcn_wmma_f32_16x16x4_f32(
        /*neg_a=*/false, a, /*neg_b=*/false, b,
        /*c_mod=*/(short)0, c, /*reuse_a=*/false, /*reuse_b=*/false);
    float s = ((c[0] + c[1]) + (c[2] + c[3])) + ((c[4] + c[5]) + (c[6] + c[7]));
    s += __shfl_xor(s, 16, 32);
    return s;
}

// Kernel 1: streaming partial reduction. Each thread handles 4 consecutive
// elements per grid-stride iteration with 128-bit non-temporal loads.
__global__ __launch_bounds__(256) void topo_partial_kernel(
    const float* __restrict__ inputs,        // (n, 2) interleaved, channel 1 is lik
    const float* __restrict__ wmap,          // (n)
    const float* __restrict__ refmap,        // (n)
    const unsigned char* __restrict__ mask,  // (n) bool as 1 byte
    float* __restrict__ partials,            // (gridDim.x * 3)
    int n)
{
    const int tid    = blockIdx.x * blockDim.x + threadIdx.x;
    const int stride = gridDim.x * blockDim.x * 4;
    const int n_vec  = n & ~3;

    float s_msq = 0.0f;  // sum(sq_err * mask)
    float s_cnt = 0.0f;  // sum(mask)
    float s_sq  = 0.0f;  // sum(sq_err)

    const v4f*          in4 = (const v4f*)inputs;
    const v4f*          w4  = (const v4f*)wmap;
    const v4f*          r4  = (const v4f*)refmap;
    const unsigned int* m4  = (const unsigned int*)mask;

    for (int base = tid * 4; base < n_vec; base += stride) {
        // elements base..base+3; inputs are (elem, channel) interleaved pairs
        v4f ia = __builtin_nontemporal_load(&in4[(base >> 1) + 0]); // e+0.c0 e+0.c1 e+1.c0 e+1.c1
        v4f ib = __builtin_nontemporal_load(&in4[(base >> 1) + 1]); // e+2.c0 e+2.c1 e+3.c0 e+3.c1
        v4f w  = __builtin_nontemporal_load(&w4[base >> 2]);
        v4f r  = __builtin_nontemporal_load(&r4[base >> 2]);
        unsigned int m = __builtin_nontemporal_load(&m4[base >> 2]);

        float lik[4] = { ia[1], ia[3], ib[1], ib[3] };
#pragma unroll
        for (int k = 0; k < 4; ++k) {
            float d  = fmaf(lik[k], w[k], -r[k]);
            float sq = d * d;
            float mf = (float)((m >> (8 * k)) & 0xFFu); // bool bytes are 0/1
            s_sq  += sq;
            s_msq += sq * mf;
            s_cnt += mf;
        }
    }

    // Scalar tail (n not multiple of 4) handled by block 0 only.
    if (blockIdx.x == 0) {
        int rem = n - n_vec;
        if ((int)threadIdx.x < rem) {
            int i = n_vec + (int)threadIdx.x;
            float d  = fmaf(inputs[2 * i + 1], wmap[i], -refmap[i]);
            float sq = d * d;
            float mf = (float)mask[i];
            s_sq += sq; s_msq += sq * mf; s_cnt += mf;
        }
    }

    // WMMA wave reduction (unconditional: full EXEC on every wave).
    float t_msq = wave_sum32(s_msq);
    float t_cnt = wave_sum32(s_cnt);
    float t_sq  = wave_sum32(s_sq);

    __shared__ float lds[3][8];
    const int lane = threadIdx.x & 31;
    const int wv   = threadIdx.x >> 5;
    if (lane == 0) { lds[0][wv] = t_msq; lds[1][wv] = t_cnt; lds[2][wv] = t_sq; }
    __syncthreads();

    if (threadIdx.x == 0) {
        float a = 0.0f, b = 0.0f, c = 0.0f;
#pragma unroll
        for (int i = 0; i < 8; ++i) { a += lds[0][i]; b += lds[1][i]; c += lds[2][i]; }
        partials[blockIdx.x * 3 + 0] = a;
        partials[blockIdx.x * 3 + 1] = b;
        partials[blockIdx.x * 3 + 2] = c;
    }
}

// Kernel 2: single 256-thread block reduces per-block partial triples and
// applies the masked/full-mean select.
__global__ __launch_bounds__(256) void topo_final_kernel(
    const float* __restrict__ partials, int nblocks, float n_total,
    float* __restrict__ out)
{
    float s_msq = 0.0f, s_cnt = 0.0f, s_sq = 0.0f;
    for (int i = threadIdx.x; i < nblocks; i += 256) {
        s_msq += partials[i * 3 + 0];
        s_cnt += partials[i * 3 + 1];
        s_sq  += partials[i * 3 + 2];
    }

    float t_msq = wave_sum32(s_msq);
    float t_cnt = wave_sum32(s_cnt);
    float t_sq  = wave_sum32(s_sq);

    __shared__ float lds[3][8];
    const int lane = threadIdx.x & 31;
    const int wv   = threadIdx.x >> 5;
    if (lane == 0) { lds[0][wv] = t_msq; lds[1][wv] = t_cnt; lds[2][wv] = t_sq; }
    __syncthreads();

    if (threadIdx.x == 0) {
        float ms = 0.0f, nm = 0.0f, fs = 0.0f;
#pragma unroll
        for (int i = 0; i < 8; ++i) { ms += lds[0][i]; nm += lds[1][i]; fs += lds[2][i]; }
        float masked_mean = ms / fmaxf(nm, 1.0f);
        float full_mean   = fs / n_total;
        out[0] = (nm > 0.0f) ? masked_mean : full_mean;
    }
}

extern "C" void kernel_launch(void* const* d_in, const int* in_sizes, int n_in,
                              void* d_out, int out_size, void* d_ws, size_t ws_size,
                              hipStream_t stream) {
    // setup_inputs order: inputs(N*2 f32), topo_cp_weight_map(N f32),
    // topo_cp_ref_map(N f32), topo_mask(N bool), height, width, batch_size
    const float*         inputs = (const float*)d_in[0];
    const float*         wmap   = (const float*)d_in[1];
    const float*         refmap = (const float*)d_in[2];
    const unsigned char* mask   = (const unsigned char*)d_in[3];
    const int n = in_sizes[1];          // B*H*W
    float* out      = (float*)d_out;
    float* partials = (float*)d_ws;

    const int threads = 256;
    int blocks = 2048;                  // 524288 threads, 8 elems/thread
    size_t need = (size_t)blocks * 3u * sizeof(float);
    if (need > ws_size) {
        blocks = (int)(ws_size / (3 * sizeof(float)));
        if (blocks < 1) blocks = 1;
        if (blocks > 2048) blocks = 2048;
    }

    topo_partial_kernel<<<blocks, threads, 0, stream>>>(inputs, wmap, refmap, mask,
                                                        partials, n);
    topo_final_kernel<<<1, threads, 0, stream>>>(partials, blocks, (float)n, out);
}